// Net_84155589197980
// MI455X (gfx1250) — compile-verified
//
#include <hip/hip_runtime.h>

typedef __attribute__((ext_vector_type(2))) float v2f;
typedef __attribute__((ext_vector_type(8))) float v8f;

#define T_STEPS 100
#define B_SZ    256
#define NI_SZ   784
#define NH_SZ   1024
#define NO_SZ   10
#define BETA_F  0.95f

#define X_ELEMS  ((size_t)T_STEPS * B_SZ * NI_SZ)   // 20,070,400
#define S1_ELEMS ((size_t)T_STEPS * B_SZ * NH_SZ)   // 26,214,400
#define S2_ELEMS ((size_t)T_STEPS * B_SZ * NO_SZ)   // 256,000

// ---------------------------------------------------------------------------
// GEMM1: cur1 = relu(X @ W1^T)
//   X  : [T*B=25600, NI=784] row-major    W1 : [NH=1024, NI=784] row-major
//   out: [T*B, NH] into the spk1 region of d_out (holds cur1 temporarily)
//
// Register-blocked: one wave32 owns a 64x64 output tile = 4x4 grid of 16x16
// v_wmma_f32_16x16x4_f32 tiles (16 accumulators, 128 acc VGPRs/lane).
// Per K-step of 4: 4 A-frag + 4 B-frag b64 loads feed 16 WMMAs
// (16 FLOP per L2 byte; 16-deep independent accumulator chains).
// Block = 8 waves covering a 256(m) x 128(n) super-tile for L1 sharing.
// ---------------------------------------------------------------------------
__global__ __launch_bounds__(256) void gemm1_wmma(const float* __restrict__ X,
                                                  const float* __restrict__ W1,
                                                  float* __restrict__ C) {
  const int lane = threadIdx.x & 31;
  const int wave = threadIdx.x >> 5;
  // grid: (25600/256) * (1024/128) = 100 * 8 = 800 blocks
  const int mBase = (blockIdx.x >> 3) * 256 + (wave >> 1) * 64;   // 4 m-waves
  const int nBase = (blockIdx.x & 7) * 128 + (wave & 1) * 64;     // 2 n-waves
  const int l15 = lane & 15;
  const int hi  = lane >> 4;

  // A frag: lane = M row, half-wave picks K pair {0,1}/{2,3} -> one b64 load.
  // B frag: lane = N col of W1^T == row of W1, same K pairing.
  const float* aRow[4];
  const float* bRow[4];
#pragma unroll
  for (int i = 0; i < 4; ++i)
    aRow[i] = X + (size_t)(mBase + 16 * i + l15) * NI_SZ + 2 * hi;
#pragma unroll
  for (int j = 0; j < 4; ++j)
    bRow[j] = W1 + (size_t)(nBase + 16 * j + l15) * NI_SZ + 2 * hi;

  v8f acc[4][4] = {};
  for (int k = 0; k < NI_SZ; k += 4) {        // 196 K-steps
    v2f a[4], b[4];
#pragma unroll
    for (int i = 0; i < 4; ++i) a[i] = *(const v2f*)(aRow[i] + k);
#pragma unroll
    for (int j = 0; j < 4; ++j) b[j] = *(const v2f*)(bRow[j] + k);
#pragma unroll
    for (int i = 0; i < 4; ++i)
#pragma unroll
      for (int j = 0; j < 4; ++j)
        acc[i][j] = __builtin_amdgcn_wmma_f32_16x16x4_f32(
            false, a[i], false, b[j], (short)0, acc[i][j], false, false);
  }

  // C/D layout: VGPR r -> row +r (lanes 0-15) / +r+8 (lanes 16-31), col = l15
#pragma unroll
  for (int i = 0; i < 4; ++i) {
#pragma unroll
    for (int j = 0; j < 4; ++j) {
      float* cBase = C + (size_t)(mBase + 16 * i + hi * 8) * NH_SZ
                       + nBase + 16 * j + l15;
#pragma unroll
      for (int r = 0; r < 8; ++r) {
        float v = acc[i][j][r];
        v = v > 0.0f ? v : 0.0f;              // relu
        cBase[(size_t)r * NH_SZ] = v;
      }
    }
  }
}

// ---------------------------------------------------------------------------
// LIF layer 1: in-place over the spk1 region. One thread per (b,h) neuron,
// scans t=0..99: read cur1[t,b,h], update membrane, overwrite with spike.
// spike = (mem - th > 0) <=> (mem > th) in fp32 (monotone rounding).
// ---------------------------------------------------------------------------
__global__ __launch_bounds__(256) void lif1_scan(float* __restrict__ buf,
                                                 const float* __restrict__ th1) {
  const int idx = blockIdx.x * blockDim.x + threadIdx.x;  // 0 .. B*NH-1
  const int h = idx & (NH_SZ - 1);
  const int b = idx >> 10;
  const float th = th1[h];
  float mem = 0.0f;
  float* p = buf + (size_t)b * NH_SZ + h;
  const size_t stride = (size_t)B_SZ * NH_SZ;
  for (int t = 0; t < T_STEPS; ++t) {
    float cur  = p[(size_t)t * stride];
    float keep = (mem > th) ? 0.0f : 1.0f;    // 1 - reset (pre-update mem)
    mem = (BETA_F * mem + cur) * keep;
    p[(size_t)t * stride] = (mem > th) ? 1.0f : 0.0f;
  }
}

// ---------------------------------------------------------------------------
// GEMM2: cur2 = relu(spk1 @ W2^T)
//   spk1: [25600, 1024]   W2: [10, 1024]   out: [25600, 10] into spk2 region.
// One wave owns 64(m) x 16(n): 4 accumulators share one masked B fragment
// (cols 10..15 zeroed arithmetically so EXEC stays all-1s through WMMAs).
// ---------------------------------------------------------------------------
__global__ __launch_bounds__(256) void gemm2_wmma(const float* __restrict__ SPK1,
                                                  const float* __restrict__ W2,
                                                  float* __restrict__ CUR2) {
  const int lane  = threadIdx.x & 31;
  const int wave  = threadIdx.x >> 5;
  const int mBase = (blockIdx.x * 8 + wave) * 64;   // 400 wave-tiles
  const int l15 = lane & 15;
  const int hi  = lane >> 4;

  const float nMask = (l15 < NO_SZ) ? 1.0f : 0.0f;
  const int   nRow  = (l15 < NO_SZ) ? l15 : 0;      // clamp: loads in-bounds

  const float* aRow[4];
#pragma unroll
  for (int i = 0; i < 4; ++i)
    aRow[i] = SPK1 + (size_t)(mBase + 16 * i + l15) * NH_SZ + 2 * hi;
  const float* bRow = W2 + (size_t)nRow * NH_SZ + 2 * hi;

  v8f acc[4] = {};
  for (int k = 0; k < NH_SZ; k += 4) {        // 256 K-steps
    v2f b = *(const v2f*)(bRow + k) * nMask;
    v2f a[4];
#pragma unroll
    for (int i = 0; i < 4; ++i) a[i] = *(const v2f*)(aRow[i] + k);
#pragma unroll
    for (int i = 0; i < 4; ++i)
      acc[i] = __builtin_amdgcn_wmma_f32_16x16x4_f32(
          false, a[i], false, b, (short)0, acc[i], false, false);
  }

  if (l15 < NO_SZ) {                          // only 10 valid output columns
#pragma unroll
    for (int i = 0; i < 4; ++i)
#pragma unroll
      for (int r = 0; r < 8; ++r) {
        float v = acc[i][r];
        v = v > 0.0f ? v : 0.0f;              // relu
        CUR2[(size_t)(mBase + 16 * i + hi * 8 + r) * NO_SZ + l15] = v;
      }
  }
}

// ---------------------------------------------------------------------------
// LIF layer 2: in-place over the spk2 region (cur2 -> spk2), mem2 recorded.
// th2 == 1.0 everywhere.
// ---------------------------------------------------------------------------
__global__ __launch_bounds__(256) void lif2_scan(float* __restrict__ buf,
                                                 float* __restrict__ mem2rec) {
  const int idx = blockIdx.x * blockDim.x + threadIdx.x;  // 0 .. B*NO-1
  if (idx >= B_SZ * NO_SZ) return;
  const int o = idx % NO_SZ;
  const int b = idx / NO_SZ;
  float mem = 0.0f;
  for (int t = 0; t < T_STEPS; ++t) {
    const size_t off = (size_t)(t * B_SZ + b) * NO_SZ + o;
    float cur  = buf[off];
    float keep = (mem > 1.0f) ? 0.0f : 1.0f;
    mem = (BETA_F * mem + cur) * keep;
    buf[off]     = (mem > 1.0f) ? 1.0f : 0.0f;
    mem2rec[off] = mem;
  }
}

// ---------------------------------------------------------------------------
extern "C" void kernel_launch(void* const* d_in, const int* in_sizes, int n_in,
                              void* d_out, int out_size, void* d_ws, size_t ws_size,
                              hipStream_t stream) {
  (void)in_sizes; (void)n_in; (void)out_size; (void)d_ws; (void)ws_size;

  const float* x   = (const float*)d_in[0];   // [T, B, NI]
  const float* w1  = (const float*)d_in[1];   // [NH, NI]
  const float* w2  = (const float*)d_in[2];   // [NO, NH]
  const float* th1 = (const float*)d_in[3];   // [NH]

  float* out      = (float*)d_out;
  float* out_x    = out;                          // T*B*NI
  float* out_spk1 = out + X_ELEMS;                // T*B*NH (cur1, then spk1)
  float* out_spk2 = out + X_ELEMS + S1_ELEMS;     // T*B*NO (cur2, then spk2)
  float* out_mem2 = out + X_ELEMS + S1_ELEMS + S2_ELEMS;  // T*B*NO

  // 1) pass-through copy of x (async d2d is graph-capture safe)
  hipMemcpyAsync(out_x, x, X_ELEMS * sizeof(float), hipMemcpyDeviceToDevice, stream);

  // 2) cur1 = relu(X @ W1^T): 800 blocks, each 8 waves covering 256x128
  gemm1_wmma<<<800, 256, 0, stream>>>(x, w1, out_spk1);

  // 3) LIF scan layer 1 (in place: cur1 -> spk1)
  lif1_scan<<<(B_SZ * NH_SZ) / 256, 256, 0, stream>>>(out_spk1, th1);

  // 4) cur2 = relu(spk1 @ W2^T): 400 wave-tiles of 64x16, 8 waves/block
  gemm2_wmma<<<50, 256, 0, stream>>>(out_spk1, w2, out_spk2);

  // 5) LIF scan layer 2 (in place: cur2 -> spk2) + record mem2
  lif2_scan<<<(B_SZ * NO_SZ + 255) / 256, 256, 0, stream>>>(out_spk2, out_mem2);
}